// CrossGATInversed_36009005809888
// MI455X (gfx1250) — compile-verified
//
#include <hip/hip_runtime.h>
#include <math.h>

#define N_NODES 100000
#define N_EDGES 1600000
#define F_IN    128
#define H_HEADS 4
#define D_HID   16
#define HD      64            // H_HEADS * D_HID
#define N_TILES 6250          // N_NODES / 16
#define ALPHA   0.2f

typedef __attribute__((ext_vector_type(2))) float v2f;
typedef __attribute__((ext_vector_type(8))) float v8f;

// ---------------- workspace layout (in floats) ----------------
// Wh   : [N, 64]            @ 0
// Bmat : [128, 64]          @ 6,400,000
// s1   : [N, H]             @ 6,408,192
// s2   : [N, H]             @ 6,808,192
// m    : [N, H]             @ 7,208,192   (float bits, atomic-max via int trick)
// denom: [N, H]             @ 7,608,192
#define WH_OFF   0
#define BM_OFF   6400000
#define S1_OFF   6408192
#define S2_OFF   6808192
#define M_OFF    7208192
#define DEN_OFF  7608192

// Reorder W[H,F,D] -> Bmat[F, H*D]  (column c = h*16+d)
__global__ void k_prep_w(const float* __restrict__ W, float* __restrict__ Bmat) {
    int i = blockIdx.x * blockDim.x + threadIdx.x;
    if (i >= F_IN * HD) return;
    int k = i / HD;
    int c = i % HD;
    int h = c / D_HID, d = c % D_HID;
    Bmat[i] = W[(h * F_IN + k) * D_HID + d];
}

// Wh = h @ Bmat + Wb via V_WMMA_F32_16X16X4_F32.
// Block = 256 threads = 8 waves; wave w owns M-tile (blockIdx.x*8 + w) and
// computes all 4 head tiles (16x16 each), reusing its A fragment 4x.
// B (32KB) is staged in LDS once per block.
__global__ void __launch_bounds__(256) k_gemm(const float* __restrict__ hin,
                                              const float* __restrict__ Bmat,
                                              const float* __restrict__ Wb,
                                              float* __restrict__ Wh) {
    __shared__ float Bs[F_IN * HD];
    for (int i = threadIdx.x; i < F_IN * HD; i += 256) Bs[i] = Bmat[i];
    __syncthreads();

    const int wave = threadIdx.x >> 5;
    const int tile = blockIdx.x * 8 + wave;
    if (tile >= N_TILES) return;          // uniform per wave: EXEC stays all-ones

    const int lane = threadIdx.x & 31;
    const int m0   = tile * 16;
    const int mrow = m0 + (lane & 15);
    const int khi  = (lane >> 4) << 1;    // 0 for lanes 0-15, 2 for lanes 16-31
    const int ncol = lane & 15;

    v8f c0 = {}, c1 = {}, c2 = {}, c3 = {};
    const float* arow = hin + (size_t)mrow * F_IN;

    for (int k0 = 0; k0 < F_IN; k0 += 4) {
        // A fragment: 16x4 f32 tile (rows m0..m0+15, K = k0..k0+3)
        v2f a;
        a.x = arow[k0 + khi];
        a.y = arow[k0 + khi + 1];
        const float* bp = &Bs[(k0 + khi) * HD + ncol];
        v2f b0, b1, b2, b3;
        b0.x = bp[0];  b0.y = bp[HD];         // head 0: cols 0..15
        b1.x = bp[16]; b1.y = bp[HD + 16];    // head 1
        b2.x = bp[32]; b2.y = bp[HD + 32];    // head 2
        b3.x = bp[48]; b3.y = bp[HD + 48];    // head 3
        c0 = __builtin_amdgcn_wmma_f32_16x16x4_f32(false, a, false, b0, (short)0, c0, false, false);
        c1 = __builtin_amdgcn_wmma_f32_16x16x4_f32(false, a, false, b1, (short)0, c1, false, false);
        c2 = __builtin_amdgcn_wmma_f32_16x16x4_f32(false, a, false, b2, (short)0, c2, false, false);
        c3 = __builtin_amdgcn_wmma_f32_16x16x4_f32(false, a, false, b3, (short)0, c3, false, false);
    }

    // C/D layout: VGPR j, lane l -> row 8*(l>>4)+j, col l&15
    const int rbase = m0 + ((lane >> 4) << 3);
    #pragma unroll
    for (int head = 0; head < 4; ++head) {
        const v8f cc = (head == 0) ? c0 : (head == 1) ? c1 : (head == 2) ? c2 : c3;
        const float bias = Wb[head * D_HID + ncol];
        #pragma unroll
        for (int j = 0; j < 8; ++j)
            Wh[(size_t)(rbase + j) * HD + head * D_HID + ncol] = cc[j] + bias;
    }
}

// Per (node, head): s1/s2 dots; init m=-inf, denom=0, out=0.
__global__ void k_node(const float* __restrict__ Wh, const float* __restrict__ a,
                       float* __restrict__ s1, float* __restrict__ s2,
                       unsigned int* __restrict__ mbits, float* __restrict__ denom,
                       float* __restrict__ out) {
    int i = blockIdx.x * blockDim.x + threadIdx.x;
    if (i >= N_NODES * H_HEADS) return;
    int n = i >> 2, hh = i & 3;
    const float* w  = Wh + (size_t)n * HD + hh * D_HID;
    const float* a1 = a + hh * 2 * D_HID;
    const float* a2 = a1 + D_HID;
    float d1 = 0.f, d2 = 0.f;
    #pragma unroll
    for (int d = 0; d < D_HID; ++d) { float v = w[d]; d1 += v * a1[d]; d2 += v * a2[d]; }
    s1[i] = d1;
    s2[i] = d2;
    mbits[i] = 0xFF800000u;  // -inf
    denom[i] = 0.f;
    float* o = out + (size_t)n * HD + hh * D_HID;
    #pragma unroll
    for (int d = 0; d < D_HID; ++d) o[d] = 0.f;
}

// ordered-bits float atomic max (always compiles; maps to L2 int atomics)
__device__ inline void atomicMaxF(float* addr, float val) {
    if (val >= 0.f) atomicMax((int*)addr, __float_as_int(val));
    else            atomicMin((unsigned int*)addr, __float_as_uint(val));
}

__global__ void k_edge_max(const int* __restrict__ src, const int* __restrict__ dst,
                           const float* __restrict__ s1, const float* __restrict__ s2,
                           const float* __restrict__ ab, float* __restrict__ m) {
    int i = blockIdx.x * blockDim.x + threadIdx.x;
    if (i >= N_EDGES * H_HEADS) return;
    int e = i >> 2, hh = i & 3;
    int s = src[e], d = dst[e];
    float ev = s1[s * H_HEADS + hh] + s2[d * H_HEADS + hh] + ab[hh];
    ev = ev > 0.f ? ev : ALPHA * ev;
    atomicMaxF(&m[d * H_HEADS + hh], ev);
}

__global__ void k_edge_accum(const int* __restrict__ src, const int* __restrict__ dst,
                             const float* __restrict__ s1, const float* __restrict__ s2,
                             const float* __restrict__ ab, const float* __restrict__ m,
                             const float* __restrict__ Wh,
                             float* __restrict__ denom, float* __restrict__ out) {
    int i = blockIdx.x * blockDim.x + threadIdx.x;
    if (i >= N_EDGES * H_HEADS) return;
    int e = i >> 2, hh = i & 3;
    int s = src[e], d = dst[e];
    float ev = s1[s * H_HEADS + hh] + s2[d * H_HEADS + hh] + ab[hh];
    ev = ev > 0.f ? ev : ALPHA * ev;
    float p = __expf(ev - m[d * H_HEADS + hh]);
    atomicAdd(&denom[d * H_HEADS + hh], p);
    const float* wsrc = Wh  + (size_t)s * HD + hh * D_HID;
    float*       o    = out + (size_t)d * HD + hh * D_HID;
    #pragma unroll
    for (int dd = 0; dd < D_HID; ++dd) atomicAdd(&o[dd], p * wsrc[dd]);
}

__global__ void k_final(const float* __restrict__ denom, float* __restrict__ out) {
    int i = blockIdx.x * blockDim.x + threadIdx.x;
    if (i >= N_NODES * H_HEADS) return;
    float inv = 1.f / fmaxf(denom[i], 1e-9f);
    float* o = out + (size_t)(i >> 2) * HD + (i & 3) * D_HID;
    #pragma unroll
    for (int dd = 0; dd < D_HID; ++dd) o[dd] *= inv;
}

extern "C" void kernel_launch(void* const* d_in, const int* in_sizes, int n_in,
                              void* d_out, int out_size, void* d_ws, size_t ws_size,
                              hipStream_t stream) {
    const float* h   = (const float*)d_in[0];
    const float* W   = (const float*)d_in[1];
    const float* Wb  = (const float*)d_in[2];
    const float* a   = (const float*)d_in[3];
    const float* ab  = (const float*)d_in[4];
    const int*   src = (const int*)d_in[5];
    const int*   dst = (const int*)d_in[6];
    float* out = (float*)d_out;
    float* ws  = (float*)d_ws;

    float* Wh    = ws + WH_OFF;
    float* Bmat  = ws + BM_OFF;
    float* s1    = ws + S1_OFF;
    float* s2    = ws + S2_OFF;
    float* m     = ws + M_OFF;
    float* denom = ws + DEN_OFF;

    k_prep_w<<<(F_IN * HD + 255) / 256, 256, 0, stream>>>(W, Bmat);
    k_gemm<<<(N_TILES + 7) / 8, 256, 0, stream>>>(h, Bmat, Wb, Wh);
    k_node<<<(N_NODES * H_HEADS + 255) / 256, 256, 0, stream>>>(
        Wh, a, s1, s2, (unsigned int*)m, denom, out);
    k_edge_max<<<(N_EDGES * H_HEADS + 255) / 256, 256, 0, stream>>>(
        src, dst, s1, s2, ab, m);
    k_edge_accum<<<(N_EDGES * H_HEADS + 255) / 256, 256, 0, stream>>>(
        src, dst, s1, s2, ab, m, Wh, denom, out);
    k_final<<<(N_NODES * H_HEADS + 255) / 256, 256, 0, stream>>>(denom, out);
}